// TopoWeightLayer_39556648796338
// MI455X (gfx1250) — compile-verified
//
#include <hip/hip_runtime.h>
#include <math.h>

// ---------------------------------------------------------------------------
// DTM layer for MI455X (gfx1250, wave32).
//
// Reformulation: the reference's top-k + cumsum + searchsorted is exactly a
// weighted quantile:  dtm = sum_{d2 < t*} d2*w  +  t* * (wb - sum_{d2<t*} w),
// where t* is the d2 value at which cumulative (ascending-distance) weight
// first reaches wb = 0.05 * sum(w). We find t* with a 4-pass 8-bit radix
// select over the IEEE-754 bits of d2 (monotonic since d2 >= 0), accumulating
// the below-threshold weight and d2*w sums during the passes. No sort needed.
//
// d2 tiles are produced by V_WMMA_F32_16X16X4_F32 using the GEMM identity:
//   A row  = [-2*x0, -2*x1, |x|^2, 1]   (16x4)
//   B col  = [ g0,    g1,   1, |g|^2]^T (4x16)
//   D      = x^2 + g^2 - 2 x.g  (exactly the reference's formula, f32)
//
// Each 256-thread workgroup (8 waves) owns 16 query rows of one batch, kept
// entirely in LDS (~306 KB of the 320 KB WGP LDS) -> selection never touches
// HBM. The weight row is streamed in with GLOBAL_LOAD_ASYNC_TO_LDS_B128
// (ASYNCcnt) overlapping the WMMA phase. All strided loops have compile-time-
// constant trip counts so EXEC stays all-1s across every WMMA.
// ---------------------------------------------------------------------------

typedef float v2f __attribute__((ext_vector_type(2)));
typedef float v8f __attribute__((ext_vector_type(8)));
typedef int   v4i __attribute__((ext_vector_type(4)));
typedef __attribute__((address_space(1))) v4i* gv4i_p;   // global int4*
typedef __attribute__((address_space(3))) v4i* lv4i_p;   // LDS int4*

#define BATCH    16
#define CHW      4096
#define QTILE    16           // queries per workgroup == WMMA M
#define THREADS  256          // 8 wave32s
#define NWAVES   (THREADS / 32)
#define NTILES   (CHW / 16)   // 256 WMMA column tiles
#define DSTRIDE  (CHW + 16)   // padded d2 row stride (bank decorrelation)
#define HSTRIDE  528          // 256 hw + 256 hdw + 16 pad
#define M0F      0.05f

// LDS layout in floats
#define OFF_D2    0
#define OFF_W     (QTILE * DSTRIDE)
#define OFF_HIST  (OFF_W + CHW)
#define OFF_STATE (OFF_HIST + QTILE * HSTRIDE)
#define LDS_FLOATS (OFF_STATE + 64)   // prefix[16] wbelow[16] dwbelow[16] wsum[1]

#if defined(__HIP_DEVICE_COMPILE__) && __has_builtin(__builtin_amdgcn_global_load_async_to_lds_b128)
#define DTM_ASYNC_W 1
#else
#define DTM_ASYNC_W 0
#endif

__global__ __launch_bounds__(THREADS)
void dtm_wmma_kernel(const float* __restrict__ X,
                     const float* __restrict__ weight,
                     const float* __restrict__ grid,
                     float* __restrict__ out)
{
    extern __shared__ float smem[];
    float*    s_d2     = smem + OFF_D2;
    float*    s_w      = smem + OFF_W;
    float*    s_hist   = smem + OFF_HIST;
    unsigned* s_prefix = (unsigned*)(smem + OFF_STATE);
    float*    s_wbelow = smem + OFF_STATE + QTILE;
    float*    s_dwbelow= smem + OFF_STATE + 2 * QTILE;
    float*    s_wsum   = smem + OFF_STATE + 3 * QTILE;

    const int tid  = threadIdx.x;
    const int lane = tid & 31;
    const int wave = tid >> 5;
    const int half = lane & 15;
    const int b     = blockIdx.x / (CHW / QTILE);
    const int qbase = (blockIdx.x % (CHW / QTILE)) * QTILE;

    if (tid == 0) s_wsum[0] = 0.0f;
    if (tid < QTILE) { s_prefix[tid] = 0u; s_wbelow[tid] = 0.0f; s_dwbelow[tid] = 0.0f; }
    __syncthreads();

    // ---- weight row -> LDS (async DMA overlapping the WMMA phase) ---------
#if DTM_ASYNC_W
#pragma unroll
    for (int j = 0; j < CHW / (THREADS * 4); ++j) {            // 4 x b128/lane
        const int i = (j * THREADS + tid) * 4;
        __builtin_amdgcn_global_load_async_to_lds_b128(
            (gv4i_p)(weight + (size_t)b * CHW + i),
            (lv4i_p)(s_w + i),
            0, 0);
    }
#else
#pragma unroll
    for (int j = 0; j < CHW / THREADS; ++j) {
        const int i = j * THREADS + tid;
        s_w[i] = weight[b * CHW + i];
    }
#endif

    // ---- A fragment: 16x4 f32. lanes 0-15 hold K=0,1; lanes 16-31 K=2,3 ---
    const int q = qbase + half;
    const float x0 = X[(b * CHW + q) * 2 + 0];
    const float x1 = X[(b * CHW + q) * 2 + 1];
    v2f afrag;
    if (lane < 16) { afrag[0] = -2.0f * x0; afrag[1] = -2.0f * x1; }
    else           { afrag[0] = x0 * x0 + x1 * x1; afrag[1] = 1.0f; }

    // ---- d2 tiles via WMMA; constant trip count => EXEC stays all-1s ------
#pragma unroll 1
    for (int it = 0; it < NTILES / NWAVES; ++it) {
        const int t = it * NWAVES + wave;
        const int n = t * 16 + half;
        const float g0 = grid[n * 2 + 0];
        const float g1 = grid[n * 2 + 1];
        v2f bfrag;   // B 4x16: lanes 0-15 K=0,1 rows; lanes 16-31 K=2,3 rows
        if (lane < 16) { bfrag[0] = g0; bfrag[1] = g1; }
        else           { bfrag[0] = 1.0f; bfrag[1] = g0 * g0 + g1 * g1; }

        v8f c = {};
        c = __builtin_amdgcn_wmma_f32_16x16x4_f32(
                /*neg_a=*/false, afrag, /*neg_b=*/false, bfrag,
                /*c_mod=*/(short)0, c, /*reuse_a=*/false, /*reuse_b=*/false);

        // D layout: lane<16 -> N=lane,   M=r   in VGPR r
        //           lane>=16-> N=lane-16, M=r+8 in VGPR r
        const int col   = t * 16 + half;
        const int rbase = (lane < 16) ? 0 : 8;
#pragma unroll
        for (int r = 0; r < 8; ++r) {
            float d = c[r];
            d = d < 0.0f ? 0.0f : d;           // reference's max(d2, 0)
            s_d2[(rbase + r) * DSTRIDE + col] = d;
        }
    }

#if DTM_ASYNC_W
#if __has_builtin(__builtin_amdgcn_s_wait_asynccnt)
    __builtin_amdgcn_s_wait_asynccnt(0);
#else
    asm volatile("s_wait_asynccnt 0" ::: "memory");
#endif
#endif
    __syncthreads();

    // ---- total mass -> wb -------------------------------------------------
    {
        float part = 0.0f;
#pragma unroll
        for (int j = 0; j < CHW / THREADS; ++j) part += s_w[j * THREADS + tid];
        atomicAdd(s_wsum, part);   // ds_add_f32
    }
    __syncthreads();
    const float wb = M0F * s_wsum[0];

    // ---- weighted radix select: 16 threads per query, 16 queries ----------
    const int g   = tid >> 4;   // query within tile
    const int tig = tid & 15;   // thread within group
    float* hw  = s_hist + g * HSTRIDE;
    float* hdw = hw + 256;
    const float* drow = s_d2 + g * DSTRIDE;

#pragma unroll 1
    for (int pass = 0; pass < 4; ++pass) {
        const int shift = 24 - 8 * pass;
        const unsigned himask = (pass == 0) ? 0u : (0xFFFFFFFFu << ((shift + 8) & 31));

#pragma unroll
        for (int jj = 0; jj < 512 / 16; ++jj) hw[jj * 16 + tig] = 0.0f;  // hw+hdw
        __syncthreads();

        const unsigned pref = s_prefix[g];
#pragma unroll 1
        for (int jj = 0; jj < CHW / 16; ++jj) {
            const int i = jj * 16 + tig;
            const float d = drow[i];
            const unsigned u = __float_as_uint(d);
            if ((u & himask) == pref) {
                const unsigned bin = (u >> shift) & 255u;
                const float wv = s_w[i];
                atomicAdd(&hw[bin],  wv);
                atomicAdd(&hdw[bin], d * wv);
            }
        }
        __syncthreads();

        if (tig == 0) {
            float acc  = s_wbelow[g];
            float accd = s_dwbelow[g];
            unsigned c = 255u;
            for (int bin = 0; bin < 256; ++bin) {
                const float hb = hw[bin];
                if (acc + hb >= wb) { c = (unsigned)bin; break; }
                acc  += hb;
                accd += hdw[bin];
            }
            s_wbelow[g]  = acc;
            s_dwbelow[g] = accd;
            s_prefix[g]  = pref | (c << shift);
        }
        __syncthreads();
    }

    // ---- finalize: dtm = dw_below + t*(wb - w_below); out = sqrt(dtm/wb) --
    if (tig == 0) {
        const float t   = __uint_as_float(s_prefix[g]);   // exact crossing d2
        const float dtm = s_dwbelow[g] + t * (wb - s_wbelow[g]);
        out[b * CHW + qbase + g] = sqrtf(dtm / wb);
    }
}

extern "C" void kernel_launch(void* const* d_in, const int* in_sizes, int n_in,
                              void* d_out, int out_size, void* d_ws, size_t ws_size,
                              hipStream_t stream) {
    (void)in_sizes; (void)n_in; (void)out_size; (void)d_ws; (void)ws_size;
    const float* X = (const float*)d_in[0];   // [B, CHW, 2]
    const float* W = (const float*)d_in[1];   // [B, CHW]
    const float* G = (const float*)d_in[2];   // [CHW, 2]
    float* out = (float*)d_out;               // [B, CHW]

    const size_t lds_bytes = (size_t)LDS_FLOATS * sizeof(float);  // ~306 KB < 320 KB WGP LDS
    (void)hipFuncSetAttribute((const void*)dtm_wmma_kernel,
                              hipFuncAttributeMaxDynamicSharedMemorySize,
                              (int)lds_bytes);

    dim3 gridDim(BATCH * (CHW / QTILE));      // 4096 workgroups
    dtm_wmma_kernel<<<gridDim, THREADS, lds_bytes, stream>>>(X, W, G, out);
}